// Mesh2Grid_Decoder_62388694942508
// MI455X (gfx1250) — compile-verified
//
#include <hip/hip_runtime.h>
#include <hip/hip_bf16.h>
#include <math.h>

// ---------------------------------------------------------------------------
// GCN mesh->grid decoder for MI455X (gfx1250, wave32).
//  * GEMMs: V_WMMA_F32_16X16X4_F32 (fp32 matrix pipe, matches fp32 reference),
//    128x64 block tile, 8 waves, double-buffered LDS tiles fed by the Tensor
//    Data Mover (tensor_load_to_lds + s_wait_tensorcnt), one barrier/K-step.
//    TDM pad_interval/pad_amount gives bank-conflict-free padded LDS rows and
//    OOB zero-fill handles the M/N tails with no clamping.
//  * Edge aggregation: global_atomic_add_f32 (msg matrix 108MB fits 192MB L2).
//  * Self-loops folded analytically into the bias/GELU pass.
// ---------------------------------------------------------------------------

typedef __attribute__((ext_vector_type(2))) float v2f;
typedef __attribute__((ext_vector_type(8))) float v8f;
typedef __attribute__((ext_vector_type(4))) unsigned int v4u;
typedef __attribute__((ext_vector_type(4))) int v4i;
typedef __attribute__((ext_vector_type(8))) int v8i;

#define NGRID  65160
#define NMESH  40962
#define NTOT   (NGRID + NMESH)   // 106122
#define NEDGE  521280
#define INCH   96
#define HIDC   256
#define OUTCH  96

#define BM 128
#define BN 64
#define BK 16
#define APAD 4                   // LDS row pad (dwords): stride 20 -> conflict-free
#define BPAD 4                   // LDS row pad (dwords): stride 68

#if __has_builtin(__builtin_amdgcn_tensor_load_to_lds)
#define USE_TDM 1
#else
#define USE_TDM 0
#endif

#if USE_TDM
// 2D tile DMA: global (row-major, stride0 elems/row) -> LDS with row padding.
// tensor_d0/d1 = remaining tensor extent from tile start (OOB zero-fills).
// pad_code: pad after 2^(pad_code+1) dwords; pad_amt_code: (amount-1) dwords.
__device__ __forceinline__ void tdm_load_2d(unsigned int lds_addr, const float* gptr,
                                            unsigned int tensor_d0, unsigned int tensor_d1,
                                            unsigned int stride0,
                                            unsigned int tile_d0, unsigned int tile_d1,
                                            unsigned int pad_code, unsigned int pad_amt_code)
{
    unsigned long long ga = (unsigned long long)(uintptr_t)gptr;
    v4u g0;
    g0.x = 1u;                                            // count=1, user descriptor
    g0.y = lds_addr;                                      // LDS byte address
    g0.z = (unsigned int)ga;                              // global_addr[31:0]
    g0.w = ((unsigned int)(ga >> 32) & 0x01FFFFFFu)       // global_addr[56:32]
         | (2u << 30);                                    // type = 2 (image)
    v8i g1;
    g1[0] = (int)((2u << 16)                              // data_size = 4B
                | (1u << 20)                              // pad_enable
                | (pad_code << 22) | (pad_amt_code << 25));
    g1[1] = (int)((tensor_d0 & 0xFFFFu) << 16);           // abar=0 | tensor_dim0 lo16
    g1[2] = (int)((tensor_d0 >> 16) | ((tensor_d1 & 0xFFFFu) << 16));
    g1[3] = (int)((tensor_d1 >> 16) | (tile_d0 << 16));
    g1[4] = (int)(tile_d1 & 0xFFFFu);                     // tile_dim2 = 0 (2D)
    g1[5] = (int)stride0;                                 // tensor_dim0_stride[31:0]
    g1[6] = 0;
    g1[7] = 0;
    v4i z4 = {0, 0, 0, 0};
#if __clang_major__ >= 23
    v8i z8 = {0, 0, 0, 0, 0, 0, 0, 0};
    __builtin_amdgcn_tensor_load_to_lds(g0, g1, z4, z4, z8, 0);
#else
    __builtin_amdgcn_tensor_load_to_lds(g0, g1, z4, z4, 0);
#endif
}
#endif

// ---------------------------------------------------------------------------
// Tiled fp32 WMMA GEMM:  C(MxN) = A(MxK) @ B(KxN) [+ bias(N)]
// 256 threads = 8 waves; wave grid 4(M) x 2(N); each wave: 2x2 tiles of 16x16.
// TRANS_OUT stores C[col*M + row] (fused output transpose for the last layer).
// ---------------------------------------------------------------------------
template <bool TRANS_OUT>
__global__ __launch_bounds__(256)
void gemm_wmma_kernel(const float* __restrict__ A, const float* __restrict__ B,
                      const float* __restrict__ bias, float* __restrict__ C,
                      int M, int N, int K)
{
    __shared__ __align__(16) float As[2][BM][BK + APAD];  // 20.0 KB
    __shared__ __align__(16) float Bs[2][BK][BN + BPAD];  //  8.5 KB

    const int tid  = threadIdx.x;
    const int lane = tid & 31;
    const int wave = tid >> 5;
    const int half = lane >> 4;      // 0: lanes 0-15 (K0/K1), 1: lanes 16-31 (K2/K3)
    const int ml   = lane & 15;      // M index for A frags, N index for B frags

    const int wm = (wave & 3) * 32;  // wave M offset inside block tile
    const int wn = (wave >> 2) * 32; // wave N offset inside block tile

    const int m0 = blockIdx.x * BM;
    const int n0 = blockIdx.y * BN;

    v8f acc[2][2];
    {
        v8f z = {};
        acc[0][0] = z; acc[0][1] = z; acc[1][0] = z; acc[1][1] = z;
    }

    const int nk = K / BK;           // K is always a multiple of 16 here

#if USE_TDM
    // One wave drives the TDM; TENSORcnt tracks both tile DMAs (in-order).
    auto tdm_issue = [&](int buf, int k0) {
        tdm_load_2d((unsigned int)(uintptr_t)&As[buf][0][0],
                    A + (size_t)m0 * K + k0,
                    (unsigned int)(K - k0), (unsigned int)(M - m0),   // OOB rows -> 0
                    (unsigned int)K, BK, BM,
                    /*pad: every 16 dw*/ 3, /*+4 dw*/ 3);
        tdm_load_2d((unsigned int)(uintptr_t)&Bs[buf][0][0],
                    B + (size_t)k0 * N + n0,
                    (unsigned int)(N - n0), (unsigned int)(K - k0),   // OOB cols -> 0
                    (unsigned int)N, BN, BK,
                    /*pad: every 64 dw*/ 5, /*+4 dw*/ 3);
    };
    if (wave == 0) tdm_issue(0, 0);
#endif

    for (int it = 0; it < nk; ++it) {
        const int buf = it & 1;
#if USE_TDM
        __builtin_amdgcn_s_wait_tensorcnt(0);   // no-op for waves 1..7
        __syncthreads();                        // tile[it] visible; buf^1 free
        if (wave == 0 && it + 1 < nk) tdm_issue(buf ^ 1, (it + 1) * BK);
#else
        const int k0 = it * BK;
        __syncthreads();
        {   // A tile 128x16: 8 floats/thread, clamped rows
            int r  = tid >> 1;
            int kq = (tid & 1) * 8;
            int grow = m0 + r; if (grow > M - 1) grow = M - 1;
            const float* ap = A + (size_t)grow * K + (k0 + kq);
#pragma unroll
            for (int j = 0; j < 8; ++j) As[buf][r][kq + j] = ap[j];
        }
        {   // B tile 16x64: 4 floats/thread, clamped cols
            int kr = tid >> 4;
            int nc = (tid & 15) * 4;
            const float* bp = B + (size_t)(k0 + kr) * N;
#pragma unroll
            for (int j = 0; j < 4; ++j) {
                int col = n0 + nc + j; if (col > N - 1) col = N - 1;
                Bs[buf][kr][nc + j] = bp[col];
            }
        }
        __syncthreads();
#endif

#pragma unroll
        for (int kk = 0; kk < BK; kk += 4) {
            // A frag 16x4: lane m=ml, VGPR0/1 = K(kk + 2*half + 0/1)
            v2f afr[2];
#pragma unroll
            for (int i = 0; i < 2; ++i) {
                int row = wm + i * 16 + ml;
                afr[i].x = As[buf][row][kk + half * 2 + 0];
                afr[i].y = As[buf][row][kk + half * 2 + 1];
            }
            // B frag 4x16: lane n=ml, VGPR0/1 = K(kk + 2*half + 0/1)
            v2f bfr[2];
#pragma unroll
            for (int j = 0; j < 2; ++j) {
                int col = wn + j * 16 + ml;
                bfr[j].x = Bs[buf][kk + half * 2 + 0][col];
                bfr[j].y = Bs[buf][kk + half * 2 + 1][col];
            }
#pragma unroll
            for (int i = 0; i < 2; ++i)
#pragma unroll
                for (int j = 0; j < 2; ++j)
                    acc[i][j] = __builtin_amdgcn_wmma_f32_16x16x4_f32(
                        false, afr[i], false, bfr[j], (short)0, acc[i][j],
                        false, false);
        }
    }

    // Store: C/D layout: VGPR r -> (M = r + 8*half, N = ml)
#pragma unroll
    for (int i = 0; i < 2; ++i) {
#pragma unroll
        for (int j = 0; j < 2; ++j) {
            int col = n0 + wn + j * 16 + ml;
            float bv = (bias != nullptr && col < N) ? bias[col] : 0.0f;
#pragma unroll
            for (int r = 0; r < 8; ++r) {
                int row = m0 + wm + i * 16 + half * 8 + r;
                if (row < M && col < N) {
                    float v = acc[i][j][r] + bv;
                    if (TRANS_OUT) C[(size_t)col * M + row] = v;
                    else           C[(size_t)row * N + col] = v;
                }
            }
        }
    }
}

// ---------------------------------------------------------------------------
// Elementwise / graph kernels
// ---------------------------------------------------------------------------
__global__ void set_ones_kernel(float* __restrict__ p, int n)
{
    int i = blockIdx.x * 256 + threadIdx.x;
    if (i < n) p[i] = 1.0f;   // self-loop contributes +1 to every degree
}

__global__ void deg_accum_kernel(const int* __restrict__ dst, float* __restrict__ deg, int e)
{
    int i = blockIdx.x * 256 + threadIdx.x;
    if (i < e) atomicAdd(&deg[dst[i]], 1.0f);
}

__global__ void rsqrt_kernel(float* __restrict__ p, int n)
{
    int i = blockIdx.x * 256 + threadIdx.x;
    if (i < n) p[i] = rsqrtf(p[i]);   // deg >= 1 always (self-loops)
}

// h[n][c] = concat(x_res_grid, x) transposed to node-major (NTOT x 96)
__global__ void concat_transpose_kernel(const float* __restrict__ xg,
                                        const float* __restrict__ xm,
                                        float* __restrict__ h)
{
    int n = blockIdx.x * 256 + threadIdx.x;
    int c = blockIdx.y;
    if (n < NTOT) {
        float v = (n < NGRID) ? xg[(size_t)c * NGRID + n]
                              : xm[(size_t)c * NMESH + (n - NGRID)];
        h[(size_t)n * INCH + c] = v;
    }
}

// One wave per edge: agg[dst] += m[src] * dinv[src]*dinv[dst]  (256 channels)
__global__ __launch_bounds__(256)
void scatter_kernel(const int* __restrict__ src, const int* __restrict__ dst,
                    const float* __restrict__ dinv, const float* __restrict__ m,
                    float* __restrict__ agg, int e)
{
    int warp = (int)((blockIdx.x * 256u + threadIdx.x) >> 5);
    int lane = threadIdx.x & 31;
    if (warp >= e) return;
    int s = src[warp], d = dst[warp];
    float w = dinv[s] * dinv[d];
    const float* mp = m + (size_t)s * HIDC;
    float*       ap = agg + (size_t)d * HIDC;
#pragma unroll
    for (int i = 0; i < HIDC / 32; ++i)
        atomicAdd(&ap[lane + i * 32], mp[lane + i * 32] * w);
}

// a[i][c] = act( a[i][c] + m[i][c]*dinv[i]^2 + bias[c] )  (self-loop + bias)
template <bool GELU>
__global__ void fuse_kernel(float* __restrict__ a, const float* __restrict__ m,
                            const float* __restrict__ dinv,
                            const float* __restrict__ bias)
{
    size_t idx = (size_t)blockIdx.x * 256 + threadIdx.x;
    if (idx >= (size_t)NTOT * HIDC) return;
    int node = (int)(idx >> 8);
    int c    = (int)(idx & 255);
    float di = dinv[node];
    float v  = a[idx] + m[idx] * di * di + bias[c];
    if (GELU) v = 0.5f * v * (1.0f + erff(v * 0.70710678118654752f));
    a[idx] = v;
}

// ---------------------------------------------------------------------------
extern "C" void kernel_launch(void* const* d_in, const int* in_sizes, int n_in,
                              void* d_out, int out_size, void* d_ws, size_t ws_size,
                              hipStream_t stream)
{
    const float* x   = (const float*)d_in[0];   // (96, 40962)
    const float* xg  = (const float*)d_in[1];   // (96, 65160)
    const int*   ei  = (const int*)d_in[2];     // (2, E) int32 (JAX default x64 off)
    const float* W1  = (const float*)d_in[3];   // (96, 256)
    const float* b1  = (const float*)d_in[4];
    const float* W2  = (const float*)d_in[5];   // (256, 256)
    const float* b2  = (const float*)d_in[6];
    const float* Wl1 = (const float*)d_in[7];   // (256, 256)
    const float* bl1 = (const float*)d_in[8];
    const float* Wl2 = (const float*)d_in[9];   // (256, 96)
    const float* bl2 = (const float*)d_in[10];
    float* out = (float*)d_out;                 // (96, 65160)

    // Workspace layout (floats): dinv | h | mbuf | abuf  (~259 MB total)
    float* ws   = (float*)d_ws;
    float* dinv = ws;
    float* h    = dinv + 106496;                        // NTOT x 96
    float* mbuf = h    + (size_t)NTOT * INCH;           // NTOT x 256  (m1 -> m2 -> t)
    float* abuf = mbuf + (size_t)NTOT * HIDC;           // NTOT x 256  (agg1/h2 -> agg2/h3)

    const int* esrc = ei;
    const int* edst = ei + NEDGE;

    // 1) degree -> dinv = rsqrt(deg), deg init 1.0 (self loops)
    set_ones_kernel<<<(NTOT + 255) / 256, 256, 0, stream>>>(dinv, NTOT);
    deg_accum_kernel<<<(NEDGE + 255) / 256, 256, 0, stream>>>(edst, dinv, NEDGE);
    rsqrt_kernel<<<(NTOT + 255) / 256, 256, 0, stream>>>(dinv, NTOT);

    // 2) node features h = concat(xg, x)^T  (NTOT x 96)
    dim3 gct((NTOT + 255) / 256, INCH);
    concat_transpose_kernel<<<gct, 256, 0, stream>>>(xg, x, h);

    // 3) GCN conv 1: m1 = h @ W1
    dim3 g1((NTOT + BM - 1) / BM, (HIDC + BN - 1) / BN);
    gemm_wmma_kernel<false><<<g1, 256, 0, stream>>>(h, W1, nullptr, mbuf, NTOT, HIDC, INCH);
    hipMemsetAsync(abuf, 0, (size_t)NTOT * HIDC * sizeof(float), stream);
    scatter_kernel<<<(NEDGE + 7) / 8, 256, 0, stream>>>(esrc, edst, dinv, mbuf, abuf, NEDGE);
    int fblk = (int)(((size_t)NTOT * HIDC + 255) / 256);
    fuse_kernel<true><<<fblk, 256, 0, stream>>>(abuf, mbuf, dinv, b1);   // + self loop + b1, GELU

    // 4) GCN conv 2: m2 = h2 @ W2  (h2 in abuf; m2 overwrites m1)
    gemm_wmma_kernel<false><<<g1, 256, 0, stream>>>(abuf, W2, nullptr, mbuf, NTOT, HIDC, HIDC);
    hipMemsetAsync(abuf, 0, (size_t)NTOT * HIDC * sizeof(float), stream);
    scatter_kernel<<<(NEDGE + 7) / 8, 256, 0, stream>>>(esrc, edst, dinv, mbuf, abuf, NEDGE);
    fuse_kernel<false><<<fblk, 256, 0, stream>>>(abuf, mbuf, dinv, b2);  // + self loop + b2

    // 5) head: t = g @ Wl1 + bl1  (g = first NGRID rows of abuf; t overwrites mbuf)
    dim3 g3((NGRID + BM - 1) / BM, (HIDC + BN - 1) / BN);
    gemm_wmma_kernel<false><<<g3, 256, 0, stream>>>(abuf, Wl1, bl1, mbuf, NGRID, HIDC, HIDC);

    // 6) out^T = t @ Wl2 + bl2, stored transposed as (96, NGRID)
    dim3 g4((NGRID + BM - 1) / BM, (OUTCH + BN - 1) / BN);
    gemm_wmma_kernel<true><<<g4, 256, 0, stream>>>(mbuf, Wl2, bl2, out, NGRID, OUTCH, HIDC);
}